// LigerCrossEntropyLoss_7370163880333
// MI455X (gfx1250) — compile-verified
//
#include <hip/hip_runtime.h>
#include <hip/hip_bf16.h>
#include <stdint.h>

// ---------------------------------------------------------------------------
// Fused softcapped cross-entropy (Liger-style) for MI455X / gfx1250.
// Memory-bound streaming reduction: 1.65 GB @ 23.3 TB/s ~= 71 us floor.
// CDNA5 async global->LDS copies (3-deep ring, non-temporal) give a
// VGPR-free prefetch pipeline; wave32 shuffles + native trans ops do the math.
// ---------------------------------------------------------------------------

#define BLOCK        256          // 8 wave32s per row
#define TILE_F4      1024         // float4s per tile = 4096 floats = 16 KB
#define TILE_ELEMS   (TILE_F4 * 4)
#define UNROLL       4            // b128 async issues per thread per tile
#define PIPE         3            // tiles in flight (48 KB LDS ring)
#define SOFTCAP      30.0f
#define ZSCALE       1e-4f
#define SMOOTH_EPS   0.1f
#define IGNORE_INDEX (-100)

#define LOG2E        1.4426950408889634f
#define LN2          0.6931471805599453f

// ---- small device helpers --------------------------------------------------

template <int N>
__device__ __forceinline__ void wait_async() {
#if __has_builtin(__builtin_amdgcn_s_wait_asynccnt)
  __builtin_amdgcn_s_wait_asynccnt(N);
#else
  asm volatile("s_wait_asynccnt %0" ::"n"(N) : "memory");
#endif
}

__device__ __forceinline__ float fast_exp2(float x) {
#if __has_builtin(__builtin_amdgcn_exp2f)
  return __builtin_amdgcn_exp2f(x);
#else
  return exp2f(x);
#endif
}

__device__ __forceinline__ float fast_log2(float x) {
#if __has_builtin(__builtin_amdgcn_logf)
  return __builtin_amdgcn_logf(x);
#else
  return log2f(x);
#endif
}

// s = 30 * tanh(x / 30)
__device__ __forceinline__ float softcap(float x) {
  float t = x * (1.0f / SOFTCAP);
#if __has_builtin(__builtin_amdgcn_tanhf)
  float th = __builtin_amdgcn_tanhf(t);   // v_tanh_f32 (CDNA5 trans op)
#else
  t = fminf(fmaxf(t, -10.0f), 10.0f);
  float a = fast_exp2(t * (2.0f * LOG2E));        // exp(2t)
#if __has_builtin(__builtin_amdgcn_rcpf)
  float th = (a - 1.0f) * __builtin_amdgcn_rcpf(a + 1.0f);
#else
  float th = (a - 1.0f) / (a + 1.0f);
#endif
#endif
  return SOFTCAP * th;
}

// exp(s - 30) = exp2(s*log2e - 30*log2e), safe: s in (-30, 30)
__device__ __forceinline__ float expsub(float s) {
  return fast_exp2(fmaf(s, LOG2E, -SOFTCAP * LOG2E));
}

// ---- pass 1: one block per token row ---------------------------------------

__global__ __launch_bounds__(BLOCK) void ce_row_kernel(
    const float* __restrict__ logits, const int* __restrict__ target,
    float* __restrict__ row_loss, float* __restrict__ row_z, int V) {
  __shared__ float4 buf[PIPE][TILE_F4];     // 3 x 16 KB staging ring
  __shared__ float red_e[BLOCK / 32];
  __shared__ float red_s[BLOCK / 32];

  const int row = blockIdx.x;
  const int tid = threadIdx.x;
  const float* rowp = logits + (size_t)row * (size_t)V;

  // LDS byte offsets of the staging buffers (generic ptr low 32 bits)
  unsigned ldsb[PIPE];
#pragma unroll
  for (int p = 0; p < PIPE; ++p) ldsb[p] = (unsigned)(size_t)(&buf[p][0]);

  const int ntiles = V / TILE_ELEMS;  // full 4096-element tiles

  // Issue UNROLL async b128 copies for one tile. Each lane stages its own
  // 16B chunks, so only a per-wave s_wait_asynccnt is needed (no barrier).
  // Stream is touched exactly once -> non-temporal to spare the 192MB L2.
  auto issue_tile = [&](int tile, unsigned ldsbase) {
    const unsigned gbase = (unsigned)tile * (unsigned)(TILE_F4 * 16);
#pragma unroll
    for (int u = 0; u < UNROLL; ++u) {
      unsigned slot = (unsigned)(u * BLOCK + tid) * 16u;
      unsigned voff = gbase + slot;   // byte offset within row (< 2^31)
      unsigned ladr = ldsbase + slot; // per-lane LDS dest address
      asm volatile("global_load_async_to_lds_b128 %0, %1, %2 th:TH_LOAD_NT"
                   ::"v"(ladr), "v"(voff), "s"((const void*)rowp)
                   : "memory");
    }
  };

  float acc_e = 0.0f;  // sum exp(s - 30)
  float acc_s = 0.0f;  // sum s

  if (ntiles > 0) {
    // prologue: fill the ring (PIPE-1 tiles ahead)
#pragma unroll
    for (int p = 0; p < PIPE - 1; ++p)
      if (p < ntiles) issue_tile(p, ldsb[p]);

    for (int t = 0; t < ntiles; ++t) {
      const int cur = t % PIPE;
      if (t + PIPE - 1 < ntiles) {
        issue_tile(t + PIPE - 1, ldsb[(t + PIPE - 1) % PIPE]);
        wait_async<(PIPE - 1) * UNROLL>();  // retire exactly tile t's loads
      } else if (t + 1 < ntiles) {
        wait_async<UNROLL>();
      } else {
        wait_async<0>();
      }
      const float4* bp = &buf[cur][0];
#pragma unroll
      for (int u = 0; u < UNROLL; ++u) {
        float4 v = bp[u * BLOCK + tid];
        float s0 = softcap(v.x), s1 = softcap(v.y);
        float s2 = softcap(v.z), s3 = softcap(v.w);
        acc_s += (s0 + s1) + (s2 + s3);
        acc_e += (expsub(s0) + expsub(s1)) + (expsub(s2) + expsub(s3));
      }
    }
  }

  // Tail: V mod 4096 elements via plain guarded global loads
  for (int i = ntiles * TILE_ELEMS + tid; i < V; i += BLOCK) {
    float s = softcap(rowp[i]);
    acc_s += s;
    acc_e += expsub(s);
  }

  // wave32 reduction, then cross-wave via LDS
#pragma unroll
  for (int o = 16; o > 0; o >>= 1) {
    acc_e += __shfl_xor(acc_e, o, 32);
    acc_s += __shfl_xor(acc_s, o, 32);
  }
  if ((tid & 31) == 0) {
    red_e[tid >> 5] = acc_e;
    red_s[tid >> 5] = acc_s;
  }
  __syncthreads();

  if (tid == 0) {
    float se = 0.0f, ss = 0.0f;
#pragma unroll
    for (int w = 0; w < BLOCK / 32; ++w) {
      se += red_e[w];
      ss += red_s[w];
    }
    // lse = 30 + ln(sum exp(s-30))
    float lse = SOFTCAP + LN2 * fast_log2(se);

    int tgt = target[row];
    int ct = (tgt >= 0 && tgt < V) ? tgt : 0;  // safe index (masked later)
    float s_t = softcap(rowp[ct]);

    float mean_s = ss * (1.0f / (float)V);
    float z = ZSCALE * lse * lse;
    float loss = (1.0f - SMOOTH_EPS) * (lse - s_t) +
                 SMOOTH_EPS * (lse - mean_s) + z;
    row_loss[row] = loss;
    row_z[row] = z;
  }
}

// ---- pass 2: deterministic final reduction ---------------------------------

#define RBLOCK 1024

__global__ __launch_bounds__(RBLOCK) void ce_reduce_kernel(
    const int* __restrict__ target, const float* __restrict__ row_loss,
    const float* __restrict__ row_z, float* __restrict__ out, int n) {
  __shared__ float rl[RBLOCK / 32];
  __shared__ float rz[RBLOCK / 32];
  __shared__ float rc[RBLOCK / 32];
  const int tid = threadIdx.x;

  float sl = 0.0f, sz = 0.0f, cnt = 0.0f;
  for (int i = tid; i < n; i += RBLOCK) {
    if (target[i] != IGNORE_INDEX) {
      sl += row_loss[i];
      sz += row_z[i];
      cnt += 1.0f;
    }
  }
#pragma unroll
  for (int o = 16; o > 0; o >>= 1) {
    sl += __shfl_xor(sl, o, 32);
    sz += __shfl_xor(sz, o, 32);
    cnt += __shfl_xor(cnt, o, 32);
  }
  if ((tid & 31) == 0) {
    rl[tid >> 5] = sl;
    rz[tid >> 5] = sz;
    rc[tid >> 5] = cnt;
  }
  __syncthreads();
  if (tid == 0) {
    float a = 0.0f, b = 0.0f, c = 0.0f;
#pragma unroll
    for (int w = 0; w < RBLOCK / 32; ++w) {
      a += rl[w];
      b += rz[w];
      c += rc[w];
    }
    c = fmaxf(c, 1.0f);
    out[0] = a / c;
    out[1] = b / c;
  }
}

// ---- host launcher ---------------------------------------------------------

extern "C" void kernel_launch(void* const* d_in, const int* in_sizes, int n_in,
                              void* d_out, int out_size, void* d_ws,
                              size_t ws_size, hipStream_t stream) {
  const float* logits = (const float*)d_in[0];
  const int* target = (const int*)d_in[1];
  float* out = (float*)d_out;

  const int n_tokens = in_sizes[1];
  const int vocab = in_sizes[0] / n_tokens;

  float* row_loss = (float*)d_ws;
  float* row_z = row_loss + n_tokens;

  ce_row_kernel<<<n_tokens, BLOCK, 0, stream>>>(logits, target, row_loss,
                                                row_z, vocab);
  ce_reduce_kernel<<<1, RBLOCK, 0, stream>>>(target, row_loss, row_z, out,
                                             n_tokens);
}